// Gate_58042188038166
// MI455X (gfx1250) — compile-verified
//
#include <hip/hip_runtime.h>
#include <hip/hip_bf16.h>

typedef float v2f __attribute__((ext_vector_type(2)));
typedef float v4f __attribute__((ext_vector_type(4)));
typedef float v8f __attribute__((ext_vector_type(8)));

#define DIM   4096
#define NEXP  64
#define NSEL  8
#define WPB   4   // waves per block

__device__ __forceinline__ v8f wmma4(v2f a, v2f b, v8f c) {
    // D = A(16x4 f32) x B(4x16 f32) + C(16x16 f32)
    return __builtin_amdgcn_wmma_f32_16x16x4_f32(
        /*neg_a=*/false, a, /*neg_b=*/false, b,
        /*c_mod=*/(short)0, c, /*reuse_a=*/false, /*reuse_b=*/false);
}

__global__ __launch_bounds__(WPB * 32)
void gate_topk_kernel(const float* __restrict__ x,
                      const float* __restrict__ w,
                      const float* __restrict__ bias,
                      float* __restrict__ probs_out,
                      float* __restrict__ idx_out,
                      int tokens)
{
    __shared__ float sc[WPB][16][NEXP + 1];   // +1 pad: row stride 65 -> conflict-free columns

    const int lane = threadIdx.x & 31;
    const int wave = threadIdx.x >> 5;
    const int ttile = blockIdx.x * WPB + wave;
    const int tok0  = ttile * 16;
    if (tok0 >= tokens) return;                // wave-uniform

    const int half = lane >> 4;                // 0 or 1
    const int sub  = lane & 15;                // row/col within tile

    // Per-lane contiguous float4 streams; (lane,vgpr)->k map identical for A and B.
    const float* arow  = x + (size_t)(tok0 + sub) * DIM + 4 * half;
    const float* brow0 = w + (size_t)(sub +  0)  * DIM + 4 * half;
    const float* brow1 = w + (size_t)(sub + 16)  * DIM + 4 * half;
    const float* brow2 = w + (size_t)(sub + 32)  * DIM + 4 * half;
    const float* brow3 = w + (size_t)(sub + 48)  * DIM + 4 * half;

    v8f acc0 = {0.f,0.f,0.f,0.f,0.f,0.f,0.f,0.f};
    v8f acc1 = acc0, acc2 = acc0, acc3 = acc0;

    #pragma unroll 4
    for (int k = 0; k < DIM; k += 8) {
        v4f a  = *(const v4f*)(arow  + k);
        v4f b0 = *(const v4f*)(brow0 + k);
        v4f b1 = *(const v4f*)(brow1 + k);
        v4f b2 = *(const v4f*)(brow2 + k);
        v4f b3 = *(const v4f*)(brow3 + k);

        v2f alo = { a.x, a.y }, ahi = { a.z, a.w };

        acc0 = wmma4(alo, (v2f){ b0.x, b0.y }, acc0);
        acc1 = wmma4(alo, (v2f){ b1.x, b1.y }, acc1);
        acc2 = wmma4(alo, (v2f){ b2.x, b2.y }, acc2);
        acc3 = wmma4(alo, (v2f){ b3.x, b3.y }, acc3);

        acc0 = wmma4(ahi, (v2f){ b0.z, b0.w }, acc0);
        acc1 = wmma4(ahi, (v2f){ b1.z, b1.w }, acc1);
        acc2 = wmma4(ahi, (v2f){ b2.z, b2.w }, acc2);
        acc3 = wmma4(ahi, (v2f){ b3.z, b3.w }, acc3);
    }

    // Spill scores (+bias) to LDS. C/D layout: VGPR r, lanes 0-15: M=r, N=sub;
    // lanes 16-31: M=r+8, N=sub. Expert index is lane-constant across the 8 rows.
    const float be0 = bias[sub +  0];
    const float be1 = bias[sub + 16];
    const float be2 = bias[sub + 32];
    const float be3 = bias[sub + 48];
    #pragma unroll
    for (int r = 0; r < 8; ++r) {
        const int m = r + 8 * half;
        sc[wave][m][sub +  0] = acc0[r] + be0;
        sc[wave][m][sub + 16] = acc1[r] + be1;
        sc[wave][m][sub + 32] = acc2[r] + be2;
        sc[wave][m][sub + 48] = acc3[r] + be3;
    }
    __syncthreads();

    // Epilogue: lanes 0-15 each own one token -> softmax + top-8.
    if (lane < 16) {
        const int tok = tok0 + lane;
        float* row = &sc[wave][lane][0];

        float mx = -INFINITY;
        for (int e = 0; e < NEXP; ++e) mx = fmaxf(mx, row[e]);

        float sum = 0.f;
        for (int e = 0; e < NEXP; ++e) {
            float v = __expf(row[e] - mx);
            row[e] = v;
            sum += v;
        }
        const float inv = 1.0f / sum;

        float* pout = probs_out + (size_t)tok * NEXP;
        for (int e = 0; e < NEXP; ++e) {
            float p = row[e] * inv;
            row[e] = p;
            pout[e] = p;
        }

        // Iterative argmax x8; strict '>' with ascending scan matches top_k tie order.
        unsigned long long used = 0ull;
        float* iout = idx_out + (size_t)tok * NSEL;
        for (int p = 0; p < NSEL; ++p) {
            float bv = -1.0f;
            int   bi = 0;
            for (int e = 0; e < NEXP; ++e) {
                float v  = row[e];
                bool  ok = (((used >> e) & 1ull) == 0ull) && (v > bv);
                bv = ok ? v : bv;
                bi = ok ? e : bi;
            }
            used |= (1ull << bi);
            iout[p] = (float)bi;
        }
    }
}

extern "C" void kernel_launch(void* const* d_in, const int* in_sizes, int n_in,
                              void* d_out, int out_size, void* d_ws, size_t ws_size,
                              hipStream_t stream) {
    const float* x    = (const float*)d_in[0];
    const float* w    = (const float*)d_in[1];
    const float* bias = (const float*)d_in[2];

    const int tokens = in_sizes[0] / DIM;      // 8192

    float* probs = (float*)d_out;                           // [tokens, 64]
    float* idx   = (float*)d_out + (size_t)tokens * NEXP;   // [tokens, 8] as float

    const int tiles  = (tokens + 15) / 16;
    const int blocks = (tiles + WPB - 1) / WPB;
    gate_topk_kernel<<<blocks, WPB * 32, 0, stream>>>(x, w, bias, probs, idx, tokens);
}